// Model_11888469475813
// MI455X (gfx1250) — compile-verified
//
#include <hip/hip_runtime.h>
#include <hip/hip_bf16.h>
#include <math.h>

typedef __attribute__((ext_vector_type(2))) float v2f;
typedef __attribute__((ext_vector_type(8))) float v8f;

#define HS_LAMBDA 0.75f

// Problem constants (match reference setup_inputs)
// x: [32,3,128,128], conv_w: [256,3,3,3], conv_b: [256], fc_w: [4096,256], fc_b: [4096]
// out: [32,4096] fp32

// ---------------------------------------------------------------------------
// Kernel 1: fused conv3x3 (as GEMM via WMMA f32 16x16x4) + bias + relu +
// hardshrink + partial pooling. Deterministic (no atomics): each workgroup
// writes its own 256-channel partial sum.
//
// Workgroup = 256 threads (8 waves). Each WG handles 4 rows of one image
// (512 pixels) as 8 pixel-blocks of 64. GEMM: M=256 channels, K=28 (27 real),
// N=64 per block. Wave w owns M-tiles {2w, 2w+1}; all 4 N-tiles per block.
// Bias is folded into the WMMA accumulator init; the two M-tile WMMA chains
// are interleaved so each dependent WMMA pair has an independent op between.
// ---------------------------------------------------------------------------
__global__ __launch_bounds__(256)
void conv_pool_wmma(const float* __restrict__ x,
                    const float* __restrict__ cw,
                    const float* __restrict__ cb,
                    float* __restrict__ partials) {
  __shared__ float sW[256][28];   // weights  [M][K], 28 KB
  __shared__ float sX[28][65];    // im2col   [K][N(+pad)], ~7.1 KB

  const int tid = threadIdx.x;
  const int g   = blockIdx.x;     // 0..1023
  const int b   = g >> 5;         // image index (32 WGs per image)
  const int blk = g & 31;
  const int h0  = blk * 4;        // 4 rows per WG

  // --- stage conv weights into LDS as [256][28], K = ci*9 + kh*3 + kw ---
  for (int i = tid; i < 256 * 28; i += 256) {
    const int m = i / 28, k = i % 28;
    sW[m][k] = (k < 27) ? cw[m * 27 + k] : 0.0f;
  }
  __syncthreads();

  const int l      = tid & 31;
  const int wv     = tid >> 5;          // wave 0..7
  const int lane16 = l & 15;
  const int hi     = l >> 4;            // 0: lanes 0-15, 1: lanes 16-31
  const int koff   = hi * 2;            // A/B frag K sub-offset per lane-half

  // --- pre-load A fragments (weights) into registers: reused all kernel ---
  // A 16x4 f32 layout: lanes 0-15 M=0..15 {K=0,1}; lanes 16-31 M=0..15 {K=2,3}
  v2f aF[2][7];
  v8f biasv[2];   // per-lane C-init: bias of channel (m0 + r + hi*8)
#pragma unroll
  for (int mt = 0; mt < 2; ++mt) {
    const int m0 = (wv * 2 + mt) * 16;
#pragma unroll
    for (int ks = 0; ks < 7; ++ks) {
      const int kb = ks * 4 + koff;
      aF[mt][ks].x = sW[m0 + lane16][kb];
      aF[mt][ks].y = sW[m0 + lane16][kb + 1];
    }
#pragma unroll
    for (int r = 0; r < 8; ++r)
      biasv[mt][r] = cb[m0 + hi * 8 + r];
  }

  // per-wave running channel sums (post-activation), 2 M-tiles x 8 VGPRs
  v8f sumAct[2] = {v8f{}, v8f{}};

  for (int p = 0; p < 8; ++p) {
    const int h  = h0 + (p >> 1);
    const int w0 = (p & 1) * 64;

    __syncthreads();  // previous block's sX reads done
    // --- build im2col tile [28][64] for pixels (b, h, w0..w0+63) ---
    for (int i = tid; i < 28 * 64; i += 256) {
      const int k = i >> 6, n = i & 63;
      float v = 0.0f;
      if (k < 27) {
        const int ci = k / 9, rem = k % 9;
        const int kh = rem / 3, kw = rem % 3;
        const int hh = h + kh - 1;
        const int ww = w0 + n + kw - 1;
        if (hh >= 0 && hh < 128 && ww >= 0 && ww < 128)
          v = x[((b * 3 + ci) * 128 + hh) * 128 + ww];
      }
      sX[k][n] = v;
    }
    __syncthreads();

    // prefetch next pixel-block's input window (global_prefetch_b8)
    if (p < 7 && tid < 27) {
      const int hn = h0 + ((p + 1) >> 1);
      const int wn = ((p + 1) & 1) * 64;
      const int ci  = tid / 9;
      const int rr  = (tid / 3) % 3;
      const int seg = tid % 3;
      const int hh  = hn - 1 + rr;
      int ww = wn - 1 + seg * 32;
      if (ww < 0) ww = 0;
      if (hh >= 0 && hh < 128)
        __builtin_prefetch(&x[((b * 3 + ci) * 128 + hh) * 128 + ww], 0, 3);
    }

    // --- GEMM tiles: 4 N-tiles, 2 M-tiles each, K chained over 7 WMMAs ---
    for (int nt = 0; nt < 4; ++nt) {
      const int n0 = nt * 16;
      // B 4x16 f32 layout: lanes 0-15 N=0..15 {K=0,1}; lanes 16-31 {K=2,3}
      v2f bF[7];
#pragma unroll
      for (int ks = 0; ks < 7; ++ks) {
        const int kb = ks * 4 + koff;
        bF[ks].x = sX[kb][n0 + lane16];
        bF[ks].y = sX[kb + 1][n0 + lane16];
      }
      // Two independent accumulator chains, interleaved per K-step so each
      // dependent WMMA->WMMA pair has an independent WMMA in between.
      // C initialized with per-channel bias (saves 8 VALU adds per tile).
      v8f acc0 = biasv[0];
      v8f acc1 = biasv[1];
#pragma unroll
      for (int ks = 0; ks < 7; ++ks) {
        acc0 = __builtin_amdgcn_wmma_f32_16x16x4_f32(
            false, aF[0][ks], false, bF[ks], (short)0, acc0, false, false);
        acc1 = __builtin_amdgcn_wmma_f32_16x16x4_f32(
            false, aF[1][ks], false, bF[ks], (short)0, acc1, false, false);
      }
      // D layout: VGPR r -> (M = m0+r+hi*8, N = n0+lane16)
      // relu then hardshrink(0.75): nonzero only if value > lambda
#pragma unroll
      for (int r = 0; r < 8; ++r) {
        const float a0 = acc0[r];
        const float a1 = acc1[r];
        sumAct[0][r] += (a0 > HS_LAMBDA) ? a0 : 0.0f;
        sumAct[1][r] += (a1 > HS_LAMBDA) ? a1 : 0.0f;
      }
    }
  }

  // --- reduce across the 16 pixel-lanes in each half, store partials ---
#pragma unroll
  for (int r = 0; r < 8; ++r) {
    float v0 = sumAct[0][r];
    float v1 = sumAct[1][r];
#pragma unroll
    for (int m = 1; m <= 8; m <<= 1) {
      v0 += __shfl_xor(v0, m, 32);
      v1 += __shfl_xor(v1, m, 32);
    }
    if (lane16 == 0) {
      const int ch0 = (wv * 2 + 0) * 16 + hi * 8 + r;
      const int ch1 = (wv * 2 + 1) * 16 + hi * 8 + r;
      partials[g * 256 + ch0] = v0;
      partials[g * 256 + ch1] = v1;
    }
  }
}

// ---------------------------------------------------------------------------
// Kernel 2: reduce 32 per-WG partials -> pooled[b][ch], scaled by 1/(128*128)
// ---------------------------------------------------------------------------
__global__ __launch_bounds__(256)
void reduce_pool(const float* __restrict__ partials, float* __restrict__ pooled) {
  const int b  = blockIdx.x;   // 0..31
  const int ch = threadIdx.x;  // 0..255
  float s = 0.0f;
#pragma unroll 4
  for (int j = 0; j < 32; ++j)
    s += partials[((b << 5) + j) * 256 + ch];
  pooled[b * 256 + ch] = s * (1.0f / 16384.0f);
}

// ---------------------------------------------------------------------------
// Kernel 3: out[b][d] = log_sigmoid(pooled[b] . fc_w[d] + fc_b[d])
// ---------------------------------------------------------------------------
__global__ __launch_bounds__(256)
void fc_logsigmoid(const float* __restrict__ pooled,
                   const float* __restrict__ fw,
                   const float* __restrict__ fb,
                   float* __restrict__ out) {
  __shared__ float sp[256];
  const int b = blockIdx.x >> 4;                         // 0..31
  const int d = ((blockIdx.x & 15) << 8) + threadIdx.x;  // 0..4095
  sp[threadIdx.x] = pooled[b * 256 + threadIdx.x];
  __syncthreads();

  const float4* __restrict__ wrow = (const float4*)(fw + (size_t)d * 256);
  const float4* __restrict__ pv   = (const float4*)sp;
  float s = fb[d];
#pragma unroll 8
  for (int k = 0; k < 64; ++k) {
    const float4 w = wrow[k];
    const float4 p = pv[k];
    s += w.x * p.x + w.y * p.y + w.z * p.z + w.w * p.w;
  }
  // stable log-sigmoid
  const float ls = fminf(s, 0.0f) - log1pf(__expf(-fabsf(s)));
  out[b * 4096 + d] = ls;
}

// ---------------------------------------------------------------------------
extern "C" void kernel_launch(void* const* d_in, const int* in_sizes, int n_in,
                              void* d_out, int out_size, void* d_ws, size_t ws_size,
                              hipStream_t stream) {
  (void)in_sizes; (void)n_in; (void)out_size; (void)ws_size;
  const float* x  = (const float*)d_in[0];  // [32,3,128,128]
  const float* cw = (const float*)d_in[1];  // [256,3,3,3]
  const float* cb = (const float*)d_in[2];  // [256]
  const float* fw = (const float*)d_in[3];  // [4096,256]
  const float* fb = (const float*)d_in[4];  // [4096]
  float* out = (float*)d_out;               // [32,4096]

  float* partials = (float*)d_ws;           // 1024*256 floats = 1 MB
  float* pooled   = partials + 1024 * 256;  // 32*256 floats

  conv_pool_wmma<<<1024, 256, 0, stream>>>(x, cw, cb, partials);
  reduce_pool<<<32, 256, 0, stream>>>(partials, pooled);
  fc_logsigmoid<<<512, 256, 0, stream>>>(pooled, fw, fb, out);
}